// MoERecurrentCell_50646254354619
// MI455X (gfx1250) — compile-verified
//
#include <hip/hip_runtime.h>
#include <math.h>

// Problem constants (match reference)
#define NB   8192   // batch
#define OBS  64
#define KS   32     // slots
#define DD   256
#define EE   8
#define HH   512
#define PERIOD 10

typedef float v2f __attribute__((ext_vector_type(2)));
typedef float v8f __attribute__((ext_vector_type(8)));

// ---- WMMA f32 16x16x4 panel: one A fragment feeds NT B tiles (16*NT cols).
// A element (m,k) at A[m*lda + k]; B element (k,n) at B[k*ldb + n].
// Layout per CDNA5 ISA 7.12.2: A VGPR0/1 = K{0,1} (lanes 0-15), K{2,3} (lanes 16-31);
// B rows striped across lanes within a VGPR.
template <int NT>
__device__ __forceinline__ void wmma_panel(v8f c[NT], const float* A, int lda,
                                           const float* B, int ldb, int kdim) {
  int lane = threadIdx.x & 31;
  int half = lane >> 4, l = lane & 15;
  for (int kk = 0; kk < kdim; kk += 4) {
    v2f a;
    a.x = A[l * lda + kk + half * 2 + 0];
    a.y = A[l * lda + kk + half * 2 + 1];
#pragma unroll
    for (int n = 0; n < NT; ++n) {
      v2f b;
      b.x = B[(kk + half * 2 + 0) * ldb + n * 16 + l];
      b.y = B[(kk + half * 2 + 1) * ldb + n * 16 + l];
      c[n] = __builtin_amdgcn_wmma_f32_16x16x4_f32(false, a, false, b,
                                                   (short)0, c[n], false, false);
    }
  }
}

// ---- Tensor Data Mover 1-D tile copy (load global->LDS, then store LDS->global)
#if __has_builtin(__builtin_amdgcn_tensor_load_to_lds) && \
    __has_builtin(__builtin_amdgcn_tensor_store_from_lds) && \
    __has_builtin(__builtin_amdgcn_s_wait_tensorcnt)
#define USE_TDM 1
typedef unsigned int u32x4 __attribute__((ext_vector_type(4)));
typedef int i32x4 __attribute__((ext_vector_type(4)));
typedef int i32x8 __attribute__((ext_vector_type(8)));

// D# group1 for a contiguous 1-D f32 tile of nelem elements (ISA 8.4):
//  w0: workgroup_mask=0 | data_size(=2:4B)<<16, no flags
//  w1[31:16] = tensor_dim0[15:0];  w2[15:0] = tensor_dim0[31:16]
//  w2[31:16] = tensor_dim1[15:0] (=1);  w3[31:16] = tile_dim0 (16-bit)
//  tile_dim1=0 (unused) -> 1-D;  w5 = tensor_dim0_stride[31:0]
__device__ __forceinline__ void tdm_copy_1d(unsigned lds_addr,
                                            unsigned long long gload,
                                            unsigned long long gstore,
                                            int nelem) {
  i32x8 g1;
  g1[0] = (2 << 16);
  g1[1] = (nelem & 0xffff) << 16;
  g1[2] = (((unsigned)nelem >> 16) & 0xffff) | (1 << 16);
  g1[3] = (nelem & 0xffff) << 16;
  g1[4] = 0;
  g1[5] = nelem;
  g1[6] = 0;
  g1[7] = 0;
  i32x4 z4 = {0, 0, 0, 0};
  u32x4 g0;
  g0[0] = 1u;                                   // count=1, user mode
  g0[1] = lds_addr;                             // LDS byte address
  g0[2] = (unsigned)gload;                      // global_addr[31:0]
  g0[3] = (unsigned)((gload >> 32) & 0x01ffffffull) | (2u << 30);  // [56:32], type=2
#if __clang_major__ >= 23
  i32x8 z8 = {0, 0, 0, 0, 0, 0, 0, 0};
  __builtin_amdgcn_tensor_load_to_lds(g0, g1, z4, z4, z8, 0);
#else
  __builtin_amdgcn_tensor_load_to_lds(g0, g1, z4, z4, 0);
#endif
  __builtin_amdgcn_s_wait_tensorcnt(0);         // tile resident in LDS
  u32x4 g0s = g0;
  g0s[2] = (unsigned)gstore;
  g0s[3] = (unsigned)((gstore >> 32) & 0x01ffffffull) | (2u << 30);
#if __clang_major__ >= 23
  __builtin_amdgcn_tensor_store_from_lds(g0s, g1, z4, z4, z8, 0);
#else
  __builtin_amdgcn_tensor_store_from_lds(g0s, g1, z4, z4, 0);
#endif
  // store drains via TENSORcnt; S_ENDPGM's implicit wait-idle covers it
}
#endif

// ---------------- K0: fold Wq/Wk into one [64,256] matrix; zero expert counts
__global__ void k_prep(const float* __restrict__ Wq_w, const float* __restrict__ Wq_b,
                       const float* __restrict__ Wk_w,
                       float* __restrict__ WQK, float* __restrict__ BQK,
                       int* __restrict__ counts) {
  int i = blockIdx.x * blockDim.x + threadIdx.x;   // 0..16383
  if (i < EE) counts[i] = 0;
  {
    int o = i >> 8, dp = i & 255;
    const float* wq = Wq_w + o * DD;
    const float* wk = Wk_w + dp * DD;
    float acc = 0.f;
    for (int d = 0; d < DD; ++d) acc += wq[d] * wk[d];
    WQK[o * DD + dp] = acc;                        // qt = x @ WQK
  }
  if (i < DD) {
    const float* wk = Wk_w + i * DD;
    float acc = 0.f;
    for (int d = 0; d < DD; ++d) acc += wk[d] * Wq_b[d];
    BQK[i] = acc;
  }
}

// ---------------- K1: qt[B,256] = x[B,64] @ WQK + BQK (WMMA f32, 16x64 panels)
__global__ void k_qt(const float* __restrict__ x, const float* __restrict__ WQK,
                     const float* __restrict__ BQK, float* __restrict__ QT) {
  int row0 = blockIdx.x * 16;
  int col0 = blockIdx.y * 64;
  v8f c[4] = {};
  wmma_panel<4>(c, x + (size_t)row0 * OBS, OBS, WQK + col0, DD, OBS);
  int lane = threadIdx.x & 31, half = lane >> 4, l = lane & 15;
  for (int n = 0; n < 4; ++n)
    for (int r = 0; r < 8; ++r) {
      int m = r + half * 8;
      QT[(size_t)(row0 + m) * DD + col0 + n * 16 + l] = c[n][r] + BQK[col0 + n * 16 + l];
    }
}

// ---------------- K2: fused attention per batch row.
// TDM moves S_prev[b] -> LDS and LDS -> S_new (bulk 32KB tile, TENSORcnt);
// logits = S_prev[b,k]·qt[b]/16, softmax over 32 slots, ctxS = sum attn*S_prev.
__global__ __launch_bounds__(256) void k_attn(const float* __restrict__ S_prev,
                                              const float* __restrict__ QT,
                                              float* __restrict__ S_new,
                                              float* __restrict__ CTXS) {
  __shared__ float s_tile[KS * DD];   // 32 KB
  __shared__ float s_qt[DD];
  __shared__ float s_logit[KS];
  __shared__ float s_attn[KS];
  int b = blockIdx.x;
  int tid = threadIdx.x;
#ifdef USE_TDM
  if (tid < 32) {   // wave 0 issues one TDM load + one TDM store for the block
    unsigned lds_off = (unsigned)(uintptr_t)s_tile;   // generic->LDS offset (low 32b)
    tdm_copy_1d(lds_off,
                (unsigned long long)(const void*)(S_prev + (size_t)b * KS * DD),
                (unsigned long long)(void*)(S_new + (size_t)b * KS * DD),
                KS * DD);
  }
#else
  {
    const float4* src = (const float4*)(S_prev + (size_t)b * KS * DD);
    float4* dst = (float4*)(S_new + (size_t)b * KS * DD);
    float4* tl  = (float4*)s_tile;
    for (int it = 0; it < (KS * DD / 4) / 256; ++it) {
      float4 v = src[it * 256 + tid];
      tl[it * 256 + tid] = v;
      dst[it * 256 + tid] = v;
    }
  }
#endif
  s_qt[tid] = QT[(size_t)b * DD + tid];
  __syncthreads();
  int wave = tid >> 5, lane = tid & 31;
  for (int kr = 0; kr < 4; ++kr) {
    int k = wave * 4 + kr;
    float acc = 0.f;
    for (int d = lane; d < DD; d += 32) acc += s_tile[k * DD + d] * s_qt[d];
    for (int off = 16; off > 0; off >>= 1) acc += __shfl_xor(acc, off);
    if (lane == 0) s_logit[k] = acc * 0.0625f;         // 1/sqrt(256)
  }
  __syncthreads();
  if (tid < 32) {
    float v = s_logit[tid];
    float m = v;
    for (int off = 16; off > 0; off >>= 1) m = fmaxf(m, __shfl_xor(m, off));
    float e = __expf(v - m);
    float s = e;
    for (int off = 16; off > 0; off >>= 1) s += __shfl_xor(s, off);
    s_attn[tid] = e / s;
  }
  __syncthreads();
  float acc = 0.f;
  for (int k = 0; k < KS; ++k) acc += s_attn[k] * s_tile[k * DD + tid];
  CTXS[(size_t)b * DD + tid] = acc;
}

// ---------------- K3: context = ctxS @ Wv_w (WMMA f32, 16x64 panels)
__global__ void k_ctx(const float* __restrict__ CTXS, const float* __restrict__ Wv_w,
                      float* __restrict__ ctx_out, float* __restrict__ feat_out) {
  int row0 = blockIdx.x * 16;
  int col0 = blockIdx.y * 64;
  v8f c[4] = {};
  wmma_panel<4>(c, CTXS + (size_t)row0 * DD, DD, Wv_w + col0, DD, DD);
  int lane = threadIdx.x & 31, half = lane >> 4, l = lane & 15;
  for (int n = 0; n < 4; ++n)
    for (int r = 0; r < 8; ++r) {
      int m = r + half * 8;
      float v = c[n][r];
      ctx_out[(size_t)(row0 + m) * DD + col0 + n * 16 + l] = v;
      feat_out[(size_t)(row0 + m) * (DD + OBS) + col0 + n * 16 + l] = v;
    }
}

// ---------------- K4: gate, features x-part, top-1 router + expert scatter
__global__ __launch_bounds__(256) void k_gate_route(
    const float* __restrict__ x, const float* __restrict__ ctx,
    const float* __restrict__ Wg_w, const float* __restrict__ Wg_b,
    const float* __restrict__ Wr_w, const float* __restrict__ Wr_b,
    float* __restrict__ GATE, float* __restrict__ feat_out,
    int* __restrict__ counts, int* __restrict__ rowlist,
    const int* __restrict__ tptr) {
  int wave = threadIdx.x >> 5, lane = threadIdx.x & 31;
  int b = blockIdx.x * 8 + wave;
  const float* xb = x + (size_t)b * OBS;
  const float* cb = ctx + (size_t)b * DD;
  for (int o = lane; o < OBS; o += 32)
    feat_out[(size_t)b * (DD + OBS) + DD + o] = xb[o];
  // gate = sigmoid(concat([x, context]) @ Wg_w + b)
  float acc = 0.f;
  for (int i = lane; i < OBS + DD; i += 32) {
    float v = (i < OBS) ? xb[i] : cb[i - OBS];
    acc += v * Wg_w[i];
  }
  for (int off = 16; off > 0; off >>= 1) acc += __shfl_xor(acc, off);
  if (lane == 0) GATE[b] = 1.f / (1.f + __expf(-(acc + Wg_b[0])));
  if (tptr[0] % PERIOD == 0) {
    float r[EE];
#pragma unroll
    for (int e = 0; e < EE; ++e) r[e] = 0.f;
    for (int d = lane; d < DD; d += 32) {
      float cv = cb[d];
#pragma unroll
      for (int e = 0; e < EE; ++e) r[e] += cv * Wr_w[d * EE + e];
    }
#pragma unroll
    for (int e = 0; e < EE; ++e)
      for (int off = 16; off > 0; off >>= 1) r[e] += __shfl_xor(r[e], off);
    if (lane == 0) {
      int best = 0; float bv = r[0] + Wr_b[0];
      for (int e = 1; e < EE; ++e) {
        float v = r[e] + Wr_b[e];
        if (v > bv) { bv = v; best = e; }   // first-max wins, matches argmax
      }
      int slot = atomicAdd(&counts[best], 1);
      rowlist[best * NB + slot] = b;
    }
  }
}

// ---------------- K_ww: non-MoE path write_vec = x @ Ww_w + Ww_b (WMMA f32)
__global__ void k_ww(const float* __restrict__ x, const float* __restrict__ Ww_w,
                     const float* __restrict__ Ww_b, float* __restrict__ WV,
                     const int* __restrict__ tptr) {
  if (tptr[0] % PERIOD == 0) return;   // MoE round handles write_vec
  int row0 = blockIdx.x * 16;
  int col0 = blockIdx.y * 64;
  v8f c[4] = {};
  wmma_panel<4>(c, x + (size_t)row0 * OBS, OBS, Ww_w + col0, DD, OBS);
  int lane = threadIdx.x & 31, half = lane >> 4, l = lane & 15;
  for (int n = 0; n < 4; ++n)
    for (int r = 0; r < 8; ++r) {
      int m = r + half * 8;
      WV[(size_t)(row0 + m) * DD + col0 + n * 16 + l] = c[n][r] + Ww_b[col0 + n * 16 + l];
    }
}

// ---------------- K5: expert-grouped MoE, 32-row gathered tiles, WMMA f32
#define MT 32
__global__ __launch_bounds__(512) void k_moe(
    const float* __restrict__ ctx,
    const float* __restrict__ e_w1, const float* __restrict__ e_b1,
    const float* __restrict__ e_w2, const float* __restrict__ e_b2,
    const int* __restrict__ counts, const int* __restrict__ rowlist,
    float* __restrict__ WV, const int* __restrict__ tptr) {
  if (tptr[0] % PERIOD != 0) return;
  int e   = blockIdx.x;
  int cnt = counts[e];
  int base = blockIdx.y * MT;
  if (base >= cnt) return;
  extern __shared__ unsigned char smem[];
  int*   s_rid = (int*)smem;                         // MT ints
  float* s_ctx = (float*)(smem + 128);               // MT x 256
  float* s_h   = s_ctx + MT * DD;                    // MT x 512
  int tid = threadIdx.x;
  if (tid < MT) s_rid[tid] = (base + tid < cnt) ? rowlist[e * NB + base + tid] : -1;
  __syncthreads();
  for (int i = tid; i < MT * DD; i += 512) {
    int r = i >> 8, d = i & 255;
    int rid = s_rid[r];
    s_ctx[i] = (rid >= 0) ? ctx[(size_t)rid * DD + d] : 0.f;
  }
  __syncthreads();
  const float* w1 = e_w1 + (size_t)e * DD * HH;
  const float* b1 = e_b1 + (size_t)e * HH;
  const float* w2 = e_w2 + (size_t)e * HH * DD;
  const float* b2 = e_b2 + (size_t)e * DD;
  int wave = tid >> 5, lane = tid & 31;
  int half = lane >> 4, l = lane & 15;
  // GEMM1: [32,256] @ [256,512] -> relu(h). 2 mt x 8 groups x 4 tiles = 64 tiles.
  {
    int mt  = wave >> 3;          // 0..1
    int ntg = (wave & 7) * 4;     // 0,4,...,28
    v8f c[4] = {};
    wmma_panel<4>(c, s_ctx + (mt * 16) * DD, DD, w1 + ntg * 16, HH, DD);
    for (int n = 0; n < 4; ++n)
      for (int r = 0; r < 8; ++r) {
        int m = mt * 16 + r + half * 8;
        int col = (ntg + n) * 16 + l;
        s_h[m * HH + col] = fmaxf(c[n][r] + b1[col], 0.f);
      }
  }
  __syncthreads();
  // GEMM2: [32,512] @ [512,256] -> scatter. 2 mt x 8 groups x 2 tiles = 32 tiles.
  {
    int mt  = wave >> 3;
    int ntg = (wave & 7) * 2;
    v8f c[2] = {};
    wmma_panel<2>(c, s_h + (mt * 16) * HH, HH, w2 + ntg * 16, DD, HH);
    for (int n = 0; n < 2; ++n)
      for (int r = 0; r < 8; ++r) {
        int m = mt * 16 + r + half * 8;
        int rid = s_rid[m];
        if (rid >= 0)
          WV[(size_t)rid * DD + (ntg + n) * 16 + l] = c[n][r] + b2[(ntg + n) * 16 + l];
      }
  }
}

// ---------------- K6: S_new[:, pos, :] = (1-g)*S_prev[:, pos, :] + g*write_vec
__global__ __launch_bounds__(256) void k_update(const float* __restrict__ S_prev,
                                                const float* __restrict__ GATE,
                                                const float* __restrict__ WV,
                                                float* __restrict__ S_new,
                                                const int* __restrict__ tptr) {
  int idx = blockIdx.x * blockDim.x + threadIdx.x;   // NB*DD
  int b = idx >> 8, d = idx & 255;
  int pos = tptr[0] % KS;
  float g = GATE[b];
  size_t off = (size_t)b * KS * DD + (size_t)pos * DD + d;
  S_new[off] = (1.f - g) * S_prev[off] + g * WV[(size_t)b * DD + d];
}

extern "C" void kernel_launch(void* const* d_in, const int* in_sizes, int n_in,
                              void* d_out, int out_size, void* d_ws, size_t ws_size,
                              hipStream_t stream) {
  (void)in_sizes; (void)n_in; (void)out_size; (void)ws_size;
  const float* S_prev = (const float*)d_in[0];
  const float* x      = (const float*)d_in[1];
  const float* Wq_w   = (const float*)d_in[2];
  const float* Wq_b   = (const float*)d_in[3];
  const float* Wk_w   = (const float*)d_in[4];
  const float* Wv_w   = (const float*)d_in[5];
  const float* Ww_w   = (const float*)d_in[6];
  const float* Ww_b   = (const float*)d_in[7];
  const float* Wg_w   = (const float*)d_in[8];
  const float* Wg_b   = (const float*)d_in[9];
  const float* Wr_w   = (const float*)d_in[10];
  const float* Wr_b   = (const float*)d_in[11];
  const float* e_w1   = (const float*)d_in[12];
  const float* e_b1   = (const float*)d_in[13];
  const float* e_w2   = (const float*)d_in[14];
  const float* e_b2   = (const float*)d_in[15];
  const int*   tptr   = (const int*)d_in[16];

  float* out      = (float*)d_out;
  float* S_new    = out;
  float* ctx_out  = out + (size_t)NB * KS * DD;
  float* feat_out = ctx_out + (size_t)NB * DD;

  float* wsf  = (float*)d_ws;
  float* WQK  = wsf;                        // 64*256
  float* BQK  = WQK + OBS * DD;             // 256
  float* QT   = BQK + DD;                   // NB*DD
  float* CTXS = QT + (size_t)NB * DD;       // NB*DD
  float* WV   = CTXS + (size_t)NB * DD;     // NB*DD
  float* GATE = WV + (size_t)NB * DD;       // NB
  int*  counts = (int*)(GATE + NB);         // EE
  int*  rows   = counts + EE;               // EE*NB

  k_prep<<<dim3(OBS * DD / 256), dim3(256), 0, stream>>>(Wq_w, Wq_b, Wk_w, WQK, BQK, counts);
  k_qt<<<dim3(NB / 16, DD / 64), dim3(32), 0, stream>>>(x, WQK, BQK, QT);
  k_attn<<<dim3(NB), dim3(256), 0, stream>>>(S_prev, QT, S_new, CTXS);
  k_ctx<<<dim3(NB / 16, DD / 64), dim3(32), 0, stream>>>(CTXS, Wv_w, ctx_out, feat_out);
  k_gate_route<<<dim3(NB / 8), dim3(256), 0, stream>>>(x, ctx_out, Wg_w, Wg_b, Wr_w, Wr_b,
                                                       GATE, feat_out, counts, rows, tptr);
  k_ww<<<dim3(NB / 16, DD / 64), dim3(32), 0, stream>>>(x, Ww_w, Ww_b, WV, tptr);
  size_t moe_smem = 128 + (size_t)MT * DD * 4 + (size_t)MT * HH * 4;  // ~98.4 KB
  k_moe<<<dim3(EE, NB / MT), dim3(512), moe_smem, stream>>>(ctx_out, e_w1, e_b1, e_w2, e_b2,
                                                            counts, rows, WV, tptr);
  k_update<<<dim3(NB * DD / 256), dim3(256), 0, stream>>>(S_prev, GATE, WV, S_new, tptr);
}